// HetGTAN_LW_76682346102822
// MI455X (gfx1250) — compile-verified
//
#include <hip/hip_runtime.h>

typedef __attribute__((ext_vector_type(2))) float v2f;
typedef __attribute__((ext_vector_type(8))) float v8f;

#define DHID 64

__device__ __forceinline__ float lrelu02(float v) { return v > 0.f ? v : 0.2f * v; }

// ---------------------------------------------------------------------------
// fc1: out = relu(X[N,F] @ W[64,F]^T + b), written to both x_out and h_out.
// One wave per 16-row tile, 4 column tiles of 16, K stepped by 4 with
// V_WMMA_F32_16X16X4_F32 (fp32 matrix pipe, preserves reference precision).
// Software-pipelined: group k+4's A/B fragment loads are issued before the
// group k WMMAs so the matrix ops cover the load latency.
// A layout: lane = m + 16*(k>=2), vgpr j -> k = j + 2*(lane>>4)
// B layout: lane = n + 16*half,  vgpr j -> k = j + 2*half
// C layout: vgpr v, lane l -> (m = v + 8*(l>>4), n = l&15)
// ---------------------------------------------------------------------------
__global__ void __launch_bounds__(256)
k_fc1_wmma(const float* __restrict__ X, const float* __restrict__ W,
           const float* __restrict__ bias, float* __restrict__ x_out,
           float* __restrict__ h_out, int N, int F) {
  const int lane = threadIdx.x & 31;
  const int wave = threadIdx.x >> 5;
  const int m16  = lane & 15;
  const int half = lane >> 4;
  const long row0 = ((long)blockIdx.x * 8 + wave) * 16;
  if (row0 >= N) return;
  long arow = row0 + m16; if (arow > (long)N - 1) arow = (long)N - 1;
  const float* __restrict__ xr = X + arow * (long)F;
  const int kk = 2 * half;

  v8f acc[4] = {};
  v2f a0, b0[4];
  // prologue loads (kb = 0)
  a0.x = xr[kk]; a0.y = xr[kk + 1];
#pragma unroll
  for (int t = 0; t < 4; ++t) {
    const float* wr = W + (long)(t * 16 + m16) * F + kk;
    b0[t].x = wr[0]; b0[t].y = wr[1];
  }
  for (int kb = 4; kb < F; kb += 4) {
    // issue next group's loads first
    v2f a1, b1[4];
    a1.x = xr[kb + kk]; a1.y = xr[kb + kk + 1];
#pragma unroll
    for (int t = 0; t < 4; ++t) {
      const float* wr = W + (long)(t * 16 + m16) * F + kb + kk;
      b1[t].x = wr[0]; b1[t].y = wr[1];
    }
    // math on the previous group's registers (loads already landed)
#pragma unroll
    for (int t = 0; t < 4; ++t)
      acc[t] = __builtin_amdgcn_wmma_f32_16x16x4_f32(false, a0, false, b0[t],
                                                     (short)0, acc[t], false, false);
    a0 = a1;
#pragma unroll
    for (int t = 0; t < 4; ++t) b0[t] = b1[t];
  }
  // epilogue group
#pragma unroll
  for (int t = 0; t < 4; ++t)
    acc[t] = __builtin_amdgcn_wmma_f32_16x16x4_f32(false, a0, false, b0[t],
                                                   (short)0, acc[t], false, false);

#pragma unroll
  for (int v = 0; v < 8; ++v) {
    long row = row0 + v + 8 * half;
    if (row < N) {
#pragma unroll
      for (int t = 0; t < 4; ++t) {
        int n = t * 16 + m16;
        float val = acc[t][v] + bias[n];
        val = val > 0.f ? val : 0.f;
        x_out[row * DHID + n] = val;
        h_out[row * DHID + n] = val;
      }
    }
  }
}

// ---------------------------------------------------------------------------
// Per-source-node: x1 = x.a1 ; w2 = exp(lrelu(x.a1 + x.a2)) ; div = w2 ;
// num = w2 * x   (self term fully initializes the accumulators each hop).
// One wave per node, butterfly reduction over 32 lanes (wave32).
// ---------------------------------------------------------------------------
__global__ void __launch_bounds__(256)
k_src_init(const float* __restrict__ x, const float* __restrict__ a1,
           const float* __restrict__ a2, float* __restrict__ x1,
           float* __restrict__ dv, float* __restrict__ num, int N) {
  const int lane = threadIdx.x & 31;
  const long n = (long)blockIdx.x * 8 + (threadIdx.x >> 5);
  if (n >= N) return;
  const float* xr = x + n * DHID;
  float v0 = xr[lane], v1 = xr[lane + 32];
  float d1 = v0 * a1[lane] + v1 * a1[lane + 32];
  float d2 = v0 * a2[lane] + v1 * a2[lane + 32];
  for (int off = 16; off > 0; off >>= 1) {
    d1 += __shfl_xor(d1, off, 32);
    d2 += __shfl_xor(d2, off, 32);
  }
  float w2 = expf(lrelu02(d1 + d2));
  if (lane == 0) { x1[n] = d1; dv[n] = w2; }
  num[n * DHID + lane]      = w2 * v0;
  num[n * DHID + lane + 32] = w2 * v1;
}

// Per-target-node: h1 = h.a2
__global__ void __launch_bounds__(256)
k_tgt_dot(const float* __restrict__ h, const float* __restrict__ a2,
          float* __restrict__ h1, int N) {
  const int lane = threadIdx.x & 31;
  const long n = (long)blockIdx.x * 8 + (threadIdx.x >> 5);
  if (n >= N) return;
  const float* hr = h + n * DHID;
  float d = hr[lane] * a2[lane] + hr[lane + 32] * a2[lane + 32];
  for (int off = 16; off > 0; off >>= 1) d += __shfl_xor(d, off, 32);
  if (lane == 0) h1[n] = d;
}

// ---------------------------------------------------------------------------
// Edge scatter: w1 = exp(lrelu(x1[s]+h1[t])); num[s,:] += w1*h[t,:]; div[s]+=w1
// One wave per edge (grid-stride). Lanes cover d = lane, lane+32.
// num/div/h are L2-resident (<= 51 MB each vs 192 MB L2); next edge's h row
// is prefetched (global_prefetch_b8) to hide gather latency.
// ---------------------------------------------------------------------------
__global__ void __launch_bounds__(256)
k_edge(const int* __restrict__ ei, int E, const float* __restrict__ x1,
       const float* __restrict__ h1, const float* __restrict__ h,
       float* __restrict__ num, float* __restrict__ dv) {
  const int lane = threadIdx.x & 31;
  const long wid = (long)blockIdx.x * 8 + (threadIdx.x >> 5);
  const long nw  = (long)gridDim.x * 8;
  for (long e = wid; e < E; e += nw) {
    int s = ei[e];
    int t = ei[(long)E + e];
    float w1 = expf(lrelu02(x1[s] + h1[t]));
    const float* hr = h + (long)t * DHID;
    float* nr = num + (long)s * DHID;
    atomicAdd(&nr[lane],      w1 * hr[lane]);
    atomicAdd(&nr[lane + 32], w1 * hr[lane + 32]);
    if (lane == 0) atomicAdd(&dv[s], w1);
    long e2 = e + nw;
    if (e2 < E) {
      __builtin_prefetch(h + (long)ei[(long)E + e2] * DHID, 0, 1);
    }
  }
}

// Mix two edge-type aggregates with softmax(lw) then ELU (paper).
__global__ void __launch_bounds__(256)
k_combine2(const float* __restrict__ numA, const float* __restrict__ dvA,
           const float* __restrict__ numB, const float* __restrict__ dvB,
           const float* __restrict__ lw, int hop, int ia, int ib,
           float* __restrict__ h, long total) {
  long i = (long)blockIdx.x * 256 + threadIdx.x;
  if (i >= total) return;
  long n = i >> 6;
  float la = lw[hop * 4 + ia], lb = lw[hop * 4 + ib];
  float m = fmaxf(la, lb);
  float ea = expf(la - m), eb = expf(lb - m);
  float inv = 1.f / (ea + eb);
  float v = (ea * inv) * (numA[i] / dvA[n]) + (eb * inv) * (numB[i] / dvB[n]);
  h[i] = v > 0.f ? v : expf(v) - 1.f;
}

// Single-edge-type combine: softmax of one logit = 1 -> h = elu(num/div).
__global__ void __launch_bounds__(256)
k_combine1(const float* __restrict__ num, const float* __restrict__ dv,
           float* __restrict__ h, long total) {
  long i = (long)blockIdx.x * 256 + threadIdx.x;
  if (i >= total) return;
  long n = i >> 6;
  float v = num[i] / dv[n];
  h[i] = v > 0.f ? v : expf(v) - 1.f;
}

// ---------------------------------------------------------------------------
// fc2: out[N,16] = hp[N,64] @ W[16,64]^T + b. Single 16-col tile, K=64.
// W staged in LDS (4 KB) once per block; A loads pipelined one group ahead.
// ---------------------------------------------------------------------------
__global__ void __launch_bounds__(256)
k_fc2_wmma(const float* __restrict__ X, const float* __restrict__ W,
           const float* __restrict__ bias, float* __restrict__ out, int N) {
  __shared__ float ws[16 * DHID];
  for (int i = threadIdx.x; i < 16 * DHID; i += 256) ws[i] = W[i];
  __syncthreads();
  const int lane = threadIdx.x & 31;
  const int wave = threadIdx.x >> 5;
  const int m16  = lane & 15;
  const int half = lane >> 4;
  const long row0 = ((long)blockIdx.x * 8 + wave) * 16;
  if (row0 >= N) return;
  long arow = row0 + m16; if (arow > (long)N - 1) arow = (long)N - 1;
  const float* xr = X + arow * DHID;
  const int kk = 2 * half;

  v8f acc = {};
  v2f a0; a0.x = xr[kk]; a0.y = xr[kk + 1];
  v2f b0; b0.x = ws[m16 * DHID + kk]; b0.y = ws[m16 * DHID + kk + 1];
#pragma unroll
  for (int kb = 4; kb < DHID; kb += 4) {
    v2f a1; a1.x = xr[kb + kk];              a1.y = xr[kb + kk + 1];
    v2f b1; b1.x = ws[m16 * DHID + kb + kk]; b1.y = ws[m16 * DHID + kb + kk + 1];
    acc = __builtin_amdgcn_wmma_f32_16x16x4_f32(false, a0, false, b0,
                                                (short)0, acc, false, false);
    a0 = a1; b0 = b1;
  }
  acc = __builtin_amdgcn_wmma_f32_16x16x4_f32(false, a0, false, b0,
                                              (short)0, acc, false, false);
#pragma unroll
  for (int v = 0; v < 8; ++v) {
    long row = row0 + v + 8 * half;
    if (row < N) out[row * 16 + m16] = acc[v] + bias[m16];
  }
}

extern "C" void kernel_launch(void* const* d_in, const int* in_sizes, int n_in,
                              void* d_out, int out_size, void* d_ws, size_t ws_size,
                              hipStream_t stream) {
  const float* x_paper   = (const float*)d_in[0];
  const float* x_author  = (const float*)d_in[1];
  const float* x_subject = (const float*)d_in[2];
  const float* fc1_pw = (const float*)d_in[3];
  const float* fc1_pb = (const float*)d_in[4];
  const float* fc1_aw = (const float*)d_in[5];
  const float* fc1_ab = (const float*)d_in[6];
  const float* fc1_sw = (const float*)d_in[7];
  const float* fc1_sb = (const float*)d_in[8];
  const float* attn1 = (const float*)d_in[9];
  const float* attn2 = (const float*)d_in[10];
  const float* lw    = (const float*)d_in[11];
  const float* fc2_w = (const float*)d_in[12];
  const float* fc2_b = (const float*)d_in[13];
  const int* ei_pa = (const int*)d_in[14];
  const int* ei_ap = (const int*)d_in[15];
  const int* ei_ps = (const int*)d_in[16];
  const int* ei_sp = (const int*)d_in[17];
  float* out = (float*)d_out;

  const int FP = in_sizes[3] / 64, FA = in_sizes[5] / 64, FS = in_sizes[7] / 64;
  const int NP = in_sizes[0] / FP, NA = in_sizes[1] / FA, NS = in_sizes[2] / FS;
  const int Epa = in_sizes[14] / 2, Eap = in_sizes[15] / 2;
  const int Eps = in_sizes[16] / 2, Esp = in_sizes[17] / 2;

  char* p = (char*)d_ws;
  auto alloc = [&](size_t elems) {
    float* r = (float*)p;
    p += ((elems * sizeof(float)) + 255) & ~(size_t)255;
    return r;
  };
  float* xp = alloc((size_t)NP * DHID);
  float* xa = alloc((size_t)NA * DHID);
  float* xs = alloc((size_t)NS * DHID);
  float* hp = alloc((size_t)NP * DHID);
  float* ha = alloc((size_t)NA * DHID);
  float* hs = alloc((size_t)NS * DHID);
  float* num0 = alloc((size_t)NP * DHID);  float* div0 = alloc(NP);
  float* num1 = alloc((size_t)NA * DHID);  float* div1 = alloc(NA);
  float* num2 = alloc((size_t)NP * DHID);  float* div2 = alloc(NP);
  float* num3 = alloc((size_t)NS * DHID);  float* div3 = alloc(NS);
  float* x1_0 = alloc(NP);  float* h1_0 = alloc(NA);
  float* x1_1 = alloc(NA);  float* h1_1 = alloc(NP);
  float* x1_2 = alloc(NP);  float* h1_2 = alloc(NS);
  float* x1_3 = alloc(NS);  float* h1_3 = alloc(NP);

  // input projections (x_dict and h_dict start identical)
  k_fc1_wmma<<<(NP + 127) / 128, 256, 0, stream>>>(x_paper,   fc1_pw, fc1_pb, xp, hp, NP, FP);
  k_fc1_wmma<<<(NA + 127) / 128, 256, 0, stream>>>(x_author,  fc1_aw, fc1_ab, xa, ha, NA, FA);
  k_fc1_wmma<<<(NS + 127) / 128, 256, 0, stream>>>(x_subject, fc1_sw, fc1_sb, xs, hs, NS, FS);

  for (int hop = 0; hop < 5; ++hop) {
    const float* a1 = attn1 + (size_t)hop * 4 * DHID;
    const float* a2 = attn2 + (size_t)hop * 4 * DHID;
    // et0: (paper -> author), aggregates at paper
    k_src_init<<<(NP + 7) / 8, 256, 0, stream>>>(xp, a1 + 0 * DHID, a2 + 0 * DHID, x1_0, div0, num0, NP);
    k_tgt_dot<<<(NA + 7) / 8, 256, 0, stream>>>(ha, a2 + 0 * DHID, h1_0, NA);
    k_edge<<<1024, 256, 0, stream>>>(ei_pa, Epa, x1_0, h1_0, ha, num0, div0);
    // et1: (author -> paper), aggregates at author
    k_src_init<<<(NA + 7) / 8, 256, 0, stream>>>(xa, a1 + 1 * DHID, a2 + 1 * DHID, x1_1, div1, num1, NA);
    k_tgt_dot<<<(NP + 7) / 8, 256, 0, stream>>>(hp, a2 + 1 * DHID, h1_1, NP);
    k_edge<<<1024, 256, 0, stream>>>(ei_ap, Eap, x1_1, h1_1, hp, num1, div1);
    // et2: (paper -> subject), aggregates at paper
    k_src_init<<<(NP + 7) / 8, 256, 0, stream>>>(xp, a1 + 2 * DHID, a2 + 2 * DHID, x1_2, div2, num2, NP);
    k_tgt_dot<<<(NS + 7) / 8, 256, 0, stream>>>(hs, a2 + 2 * DHID, h1_2, NS);
    k_edge<<<256, 256, 0, stream>>>(ei_ps, Eps, x1_2, h1_2, hs, num2, div2);
    // et3: (subject -> paper), aggregates at subject
    k_src_init<<<(NS + 7) / 8, 256, 0, stream>>>(xs, a1 + 3 * DHID, a2 + 3 * DHID, x1_3, div3, num3, NS);
    k_tgt_dot<<<(NP + 7) / 8, 256, 0, stream>>>(hp, a2 + 3 * DHID, h1_3, NP);
    k_edge<<<256, 256, 0, stream>>>(ei_sp, Esp, x1_3, h1_3, hp, num3, div3);
    // combine (writes h in place; num/div are separate buffers)
    long tp = (long)NP * DHID, ta = (long)NA * DHID, ts = (long)NS * DHID;
    k_combine2<<<(unsigned)((tp + 255) / 256), 256, 0, stream>>>(num0, div0, num2, div2, lw, hop, 0, 2, hp, tp);
    k_combine1<<<(unsigned)((ta + 255) / 256), 256, 0, stream>>>(num1, div1, ha, ta);
    k_combine1<<<(unsigned)((ts + 255) / 256), 256, 0, stream>>>(num3, div3, hs, ts);
  }

  // output head
  k_fc2_wmma<<<(NP + 127) / 128, 256, 0, stream>>>(hp, fc2_w, fc2_b, out, NP);
}